// DiscoveryEngineModel_76209899700424
// MI455X (gfx1250) — compile-verified
//
#include <hip/hip_runtime.h>
#include <math.h>

#define Nn 16384
#define Ee 262144
#define Bb 8
#define Kk 16
#define Cc 256
#define Ff 8
#define Ll 32
#define EPSf 1e-9f
#define DTf 0.1f

typedef __attribute__((ext_vector_type(2))) float v2f;
typedef __attribute__((ext_vector_type(8))) float v8f;

__device__ __forceinline__ void atomAddF(float* p, float v) {
    __hip_atomic_fetch_add(p, v, __ATOMIC_RELAXED, __HIP_MEMORY_SCOPE_AGENT);
}

// ---------------------------------------------------------------------------
// Weight repack: W (K x ON row-major) -> Wp in fragment order
//   Wp[ ((k4*(ON/16) + nt)*16 + r)*4 + hi*2 + j ] = W[(k4*4 + hi*2 + j)*ON + nt*16 + r]
// so each lane's B fragment is one aligned 8B load and a whole fragment is a
// contiguous, fully-coalesced 256B block.
// ---------------------------------------------------------------------------
template<int K, int ON>
__global__ __launch_bounds__(256)
void repack_kernel(const float* __restrict__ W, float* __restrict__ Wp)
{
    const int idx = blockIdx.x * 256 + threadIdx.x;
    const int j    = idx & 1;
    const int hi   = (idx >> 1) & 1;
    const int r    = (idx >> 2) & 15;
    const int rest = idx >> 6;
    const int nt   = rest % (ON / 16);
    const int k4   = rest / (ON / 16);
    const int k = k4 * 4 + hi * 2 + j;
    const int n = nt * 16 + r;
    Wp[idx] = W[k * ON + n];
}

// ---------------------------------------------------------------------------
// Wave-level fp32 WMMA tile GEMM: out[16][ON] = act(in[16][K] @ W + b)
// (W given in repacked fragment order, see repack_kernel.)
// Fragment layouts per CDNA5 ISA 7.12.2 (16x16x4 F32):
//   A (16x4):  lane m = lane&15, K pair = (lane>>4)*2 .. +1
//   B (4x16):  lane n = lane&15, K pair = (lane>>4)*2 .. +1
//   C/D:       vgpr v -> row m = v + 8*(lane>>4), col n = lane&15
// ACT: 0 = none, 1 = relu, 2 = leaky relu (0.1)
// ---------------------------------------------------------------------------
template<int K, int ON, int ACT>
__device__ __forceinline__ void waveGemmTile(const float* in0,
                                             const float* __restrict__ Wp,
                                             const float* __restrict__ bias,
                                             float* out0, int lane)
{
    const int r  = lane & 15;
    const int hi = lane >> 4;
    for (int nt = 0; nt < ON / 16; ++nt) {
        v8f c = {};
        const float* wp = Wp + ((nt * 16 + r) * 4 + hi * 2);
        for (int k4 = 0; k4 < K / 4; ++k4) {
            v2f b = *(const v2f*)(wp + (size_t)k4 * (ON / 16) * 64);
            v2f a = *(const v2f*)(in0 + r * K + k4 * 4 + hi * 2);
            c = __builtin_amdgcn_wmma_f32_16x16x4_f32(false, a, false, b,
                                                      (short)0, c, false, false);
        }
        float bv = bias[nt * 16 + r];
#pragma unroll
        for (int v = 0; v < 8; ++v) {
            float val = c[v] + bv;
            if (ACT == 1) val = fmaxf(val, 0.f);
            else if (ACT == 2) val = (val >= 0.f) ? val : 0.1f * val;
            out0[(v + 8 * hi) * ON + nt * 16 + r] = val;
        }
    }
}

// Two M-tiles sharing each B (weight) fragment: 2x arithmetic intensity on the
// L2-resident weight stream.
template<int K, int ON, int ACT>
__device__ __forceinline__ void waveGemmTile2(const float* in0, const float* in1,
                                              const float* __restrict__ Wp,
                                              const float* __restrict__ bias,
                                              float* out0, float* out1, int lane)
{
    const int r  = lane & 15;
    const int hi = lane >> 4;
    for (int nt = 0; nt < ON / 16; ++nt) {
        v8f c0 = {}, c1 = {};
        const float* wp = Wp + ((nt * 16 + r) * 4 + hi * 2);
        for (int k4 = 0; k4 < K / 4; ++k4) {
            v2f b  = *(const v2f*)(wp + (size_t)k4 * (ON / 16) * 64);
            v2f a0 = *(const v2f*)(in0 + r * K + k4 * 4 + hi * 2);
            v2f a1 = *(const v2f*)(in1 + r * K + k4 * 4 + hi * 2);
            c0 = __builtin_amdgcn_wmma_f32_16x16x4_f32(false, a0, false, b,
                                                       (short)0, c0, false, false);
            c1 = __builtin_amdgcn_wmma_f32_16x16x4_f32(false, a1, false, b,
                                                       (short)0, c1, false, false);
        }
        float bv = bias[nt * 16 + r];
#pragma unroll
        for (int v = 0; v < 8; ++v) {
            float v0 = c0[v] + bv;
            float v1 = c1[v] + bv;
            if (ACT == 1) { v0 = fmaxf(v0, 0.f); v1 = fmaxf(v1, 0.f); }
            else if (ACT == 2) {
                v0 = (v0 >= 0.f) ? v0 : 0.1f * v0;
                v1 = (v1 >= 0.f) ? v1 : 0.1f * v1;
            }
            out0[(v + 8 * hi) * ON + nt * 16 + r] = v0;
            out1[(v + 8 * hi) * ON + nt * 16 + r] = v1;
        }
    }
}

// ---------------------------------------------------------------------------
// Fused per-edge 3-layer MLP + segment_sum scatter.  IN = 2*width(h).
// One wave processes 32 edges (two 16-row M tiles sharing weight fragments).
// Dynamic LDS: sIn 32*IN, sA 32*256 [, sB 32*256 when IN < 256].
// ---------------------------------------------------------------------------
template<int IN>
__global__ __launch_bounds__(32)
void gnn_edge_kernel(const float* __restrict__ hIn, const int* __restrict__ ei,
                     const float* __restrict__ w1, const float* __restrict__ b1,
                     const float* __restrict__ w2, const float* __restrict__ b2,
                     const float* __restrict__ w3, const float* __restrict__ b3,
                     float* __restrict__ acc)
{
    constexpr int INH = IN / 2;
    extern __shared__ float smem[];
    float* sIn = smem;               // 32*IN
    float* sA  = smem + 32 * IN;     // 32*256
    float* sB  = (IN >= 256) ? sIn : (sA + 32 * 256);
    __shared__ int sDst[32];
    const int lane = threadIdx.x;
    const int e0 = blockIdx.x * 32;

    sDst[lane] = ei[Ee + e0 + lane];
    for (int i = 0; i < 32; ++i) {
        const int e = e0 + i;
        const int sN = ei[e];        // src node
        const int dN = ei[Ee + e];   // dst node
        if (IN >= 128) {
            for (int j = lane * 4; j < IN; j += 128) {
                float4 v = (j < INH) ? *(const float4*)&hIn[(size_t)dN * INH + j]
                                     : *(const float4*)&hIn[(size_t)sN * INH + (j - INH)];
                *(float4*)&sIn[i * IN + j] = v;
            }
        } else {
            for (int j = lane; j < IN; j += 32)
                sIn[i * IN + j] = (j < INH) ? hIn[(size_t)dN * INH + j]
                                            : hIn[(size_t)sN * INH + (j - INH)];
        }
    }
    __syncthreads();

    waveGemmTile2<IN, 256, 1>(sIn, sIn + 16 * IN, w1, b1, sA, sA + 16 * 256, lane);
    __syncthreads();
    waveGemmTile2<256, 256, 1>(sA, sA + 16 * 256, w2, b2, sB, sB + 16 * 256, lane);
    __syncthreads();
    waveGemmTile2<256, 256, 0>(sB, sB + 16 * 256, w3, b3, sA, sA + 16 * 256, lane);
    __syncthreads();

    for (int i = 0; i < 32; ++i) {
        const int dN = sDst[i];
        for (int cidx = lane; cidx < 256; cidx += 32)
            atomAddF(&acc[(size_t)dN * 256 + cidx], sA[i * 256 + cidx]);
    }
}

// relu + LayerNorm per node (one wave per node, b128 access)
__global__ __launch_bounds__(32)
void relu_ln_kernel(const float* __restrict__ acc, const float* __restrict__ gam,
                    const float* __restrict__ bet, float* __restrict__ out)
{
    const int n = blockIdx.x;
    const int lane = threadIdx.x;
    float4 va = *(const float4*)&acc[(size_t)n * 256 + lane * 8];
    float4 vb = *(const float4*)&acc[(size_t)n * 256 + lane * 8 + 4];
    float vals[8] = { fmaxf(va.x, 0.f), fmaxf(va.y, 0.f), fmaxf(va.z, 0.f), fmaxf(va.w, 0.f),
                      fmaxf(vb.x, 0.f), fmaxf(vb.y, 0.f), fmaxf(vb.z, 0.f), fmaxf(vb.w, 0.f) };
    float sum = 0.f, sq = 0.f;
#pragma unroll
    for (int i = 0; i < 8; ++i) { sum += vals[i]; sq += vals[i] * vals[i]; }
    for (int off = 16; off >= 1; off >>= 1) {
        sum += __shfl_xor(sum, off);
        sq  += __shfl_xor(sq, off);
    }
    const float m = sum * (1.f / 256.f);
    const float var = sq * (1.f / 256.f) - m * m;
    const float inv = rsqrtf(var + 1e-5f);
    float4 oa, ob;
    float* po = (float*)&oa;
#pragma unroll
    for (int i = 0; i < 4; ++i)
        po[i] = (vals[i] - m) * inv * gam[lane * 8 + i] + bet[lane * 8 + i];
    po = (float*)&ob;
#pragma unroll
    for (int i = 0; i < 4; ++i)
        po[i] = (vals[4 + i] - m) * inv * gam[lane * 8 + 4 + i] + bet[lane * 8 + 4 + i];
    *(float4*)&out[(size_t)n * 256 + lane * 8]     = oa;
    *(float4*)&out[(size_t)n * 256 + lane * 8 + 4] = ob;
}

// pool MLP (WMMA, 32 nodes per wave) + gumbel softmax -> s (N x 16)
__global__ __launch_bounds__(32)
void pool_softmax_kernel(const float* __restrict__ h,
                         const float* __restrict__ w1, const float* __restrict__ b1,
                         const float* __restrict__ w2, const float* __restrict__ b2,
                         const float* __restrict__ scaling, const float* __restrict__ g,
                         float* __restrict__ sOut)
{
    extern __shared__ float smem[];   // 32*256
    __shared__ float sL[32 * 16];
    const int lane = threadIdx.x;
    const int n0 = blockIdx.x * 32;

    waveGemmTile2<256, 256, 1>(h + (size_t)n0 * 256, h + (size_t)(n0 + 16) * 256,
                               w1, b1, smem, smem + 16 * 256, lane);
    __syncthreads();
    {
        const int r = lane & 15, hi = lane >> 4;
        v8f c0 = {}, c1 = {};
        const float* wp = w2 + (r * 4 + hi * 2);   // ON = 16, nt = 0
        for (int k4 = 0; k4 < 64; ++k4) {
            v2f b  = *(const v2f*)(wp + k4 * 64);
            v2f a0 = *(const v2f*)(smem + r * 256 + k4 * 4 + hi * 2);
            v2f a1 = *(const v2f*)(smem + (16 + r) * 256 + k4 * 4 + hi * 2);
            c0 = __builtin_amdgcn_wmma_f32_16x16x4_f32(false, a0, false, b,
                                                       (short)0, c0, false, false);
            c1 = __builtin_amdgcn_wmma_f32_16x16x4_f32(false, a1, false, b,
                                                       (short)0, c1, false, false);
        }
        const float sc = scaling[0];
#pragma unroll
        for (int v = 0; v < 8; ++v) {
            sL[(v + 8 * hi) * 16 + r]      = (c0[v] + b2[r]) * sc;
            sL[(16 + v + 8 * hi) * 16 + r] = (c1[v] + b2[r]) * sc;
        }
    }
    __syncthreads();
    {
        const int n = n0 + lane;
        float t[16];
        float mx = -1e30f;
        for (int k = 0; k < 16; ++k) {
            float u = sL[lane * 16 + k] + g[(size_t)n * 16 + k]; // TAU == 1.0
            t[k] = u; mx = fmaxf(mx, u);
        }
        float sum = 0.f;
        for (int k = 0; k < 16; ++k) { t[k] = expf(t[k] - mx); sum += t[k]; }
        const float inv = 1.f / sum;
        for (int k = 0; k < 16; ++k) sOut[(size_t)n * 16 + k] = t[k] * inv;
    }
}

__global__ void range_init_kernel(int* startB, int* endB)
{
    const int t = threadIdx.x;
    if (t < Bb) { startB[t] = Nn; endB[t] = 0; }
}

__global__ __launch_bounds__(256)
void range_kernel(const int* __restrict__ batch, int* startB, int* endB)
{
    const int n = blockIdx.x * 256 + threadIdx.x;
    const int b = batch[n];
    atomicMin(&startB[b], n);
    atomicMax(&endB[b], n + 1);
}

// node statistics: entropy acc, per-batch (sum s, sum s*pos), A3 = sum s*|pos|^2,
// dots = s.T @ s.  One block handles 256 nodes.
__global__ __launch_bounds__(256)
void stats_kernel(const float* __restrict__ s, const float* __restrict__ x,
                  const int* __restrict__ batch,
                  float* Sb0, float* Sb1, float* A3a, float* dots, float* entAcc)
{
    __shared__ float sS[256 * 16];
    __shared__ float sSq[256];
    __shared__ float red[256];
    const int t = threadIdx.x;
    const int n = blockIdx.x * 256 + t;

    const float px = x[(size_t)n * Ff + 0];
    const float py = x[(size_t)n * Ff + 1];
    sSq[t] = px * px + py * py;
    float ent = 0.f;
    for (int k = 0; k < 16; ++k) {
        const float v = s[(size_t)n * 16 + k];
        sS[t * 16 + k] = v;
        ent += v * logf(v + EPSf);
    }
    red[t] = ent;
    __syncthreads();
    for (int o = 128; o > 0; o >>= 1) {
        if (t < o) red[t] += red[t + o];
        __syncthreads();
    }
    if (t == 0) atomAddF(entAcc, red[0]);

    const int b = batch[n];
    for (int k = 0; k < 16; ++k) {
        const float v = sS[t * 16 + k];
        atomAddF(&Sb0[b * 16 + k], v);
        atomAddF(&Sb1[(b * 16 + k) * 2 + 0], v * px);
        atomAddF(&Sb1[(b * 16 + k) * 2 + 1], v * py);
    }
    __syncthreads();
    if (t < 16) {
        float a3 = 0.f;
        for (int i = 0; i < 256; ++i) a3 += sS[i * 16 + t] * sSq[i];
        atomAddF(&A3a[t], a3);
    }
    {
        const int k1 = t >> 4, k2 = t & 15;
        float d = 0.f;
        for (int i = 0; i < 256; ++i) d += sS[i * 16 + k1] * sS[i * 16 + k2];
        atomAddF(&dots[t], d);
    }
}

__global__ __launch_bounds__(256)
void conn_kernel(const float* __restrict__ s, const int* __restrict__ ei, float* connAcc)
{
    __shared__ float red[256];
    const int t = threadIdx.x;
    const int e = blockIdx.x * 256 + t;
    const int sN = ei[e], dN = ei[Ee + e];
    float acc = 0.f;
    for (int k = 0; k < 16; ++k) {
        const float d = s[(size_t)sN * 16 + k] - s[(size_t)dN * 16 + k];
        acc += d * d;
    }
    red[t] = acc;
    __syncthreads();
    for (int o = 128; o > 0; o >>= 1) {
        if (t < o) red[t] += red[t + o];
        __syncthreads();
    }
    if (t == 0) atomAddF(connAcc, red[0]);
}

// pooled[b] = s_b^T (16 x Nb) @ h_b (Nb x 256), WMMA over node (K) dimension
__global__ __launch_bounds__(32)
void pooled_kernel(const float* __restrict__ s, const float* __restrict__ h,
                   const int* __restrict__ startB, const int* __restrict__ endB,
                   float* __restrict__ pooled)
{
    const int b = blockIdx.x;
    const int lane = threadIdx.x;
    const int r = lane & 15, hi = lane >> 4;
    const int n0 = startB[b], n1 = endB[b];
    for (int nt = 0; nt < 16; ++nt) {
        v8f c = {};
        for (int n = n0; n < n1; n += 4) {
            const int k0 = n + hi * 2, k1 = k0 + 1;
            v2f a, bb;
            a[0]  = (k0 < n1) ? s[(size_t)k0 * 16 + r] : 0.f;
            a[1]  = (k1 < n1) ? s[(size_t)k1 * 16 + r] : 0.f;
            bb[0] = (k0 < n1) ? h[(size_t)k0 * 256 + nt * 16 + r] : 0.f;
            bb[1] = (k1 < n1) ? h[(size_t)k1 * 256 + nt * 16 + r] : 0.f;
            c = __builtin_amdgcn_wmma_f32_16x16x4_f32(false, a, false, bb,
                                                      (short)0, c, false, false);
        }
#pragma unroll
        for (int v = 0; v < 8; ++v)
            pooled[(size_t)b * 16 * 256 + (v + 8 * hi) * 256 + nt * 16 + r] = c[v];
    }
}

// enc_out: (B*K) x 256 -> relu(@w1) @ w2 + b2 -> latent (B*K) x 32
__global__ __launch_bounds__(32)
void encout_kernel(const float* __restrict__ pooled,
                   const float* __restrict__ w1, const float* __restrict__ b1,
                   const float* __restrict__ w2, const float* __restrict__ b2,
                   float* __restrict__ latent)
{
    __shared__ float sA[16 * 256];
    const int lane = threadIdx.x;
    const int row0 = blockIdx.x * 16;
    waveGemmTile<256, 256, 1>(pooled + (size_t)row0 * 256, w1, b1, sA, lane);
    __syncthreads();
    const int r = lane & 15, hi = lane >> 4;
    for (int nt = 0; nt < 2; ++nt) {
        v8f c = {};
        const float* wp = w2 + (((nt) * 16 + r) * 4 + hi * 2);  // ON = 32
        for (int k4 = 0; k4 < 64; ++k4) {
            v2f b = *(const v2f*)(wp + k4 * 2 * 64);
            v2f a = *(const v2f*)(sA + r * 256 + k4 * 4 + hi * 2);
            c = __builtin_amdgcn_wmma_f32_16x16x4_f32(false, a, false, b,
                                                      (short)0, c, false, false);
        }
#pragma unroll
        for (int v = 0; v < 8; ++v)
            latent[(size_t)(row0 + v + 8 * hi) * 32 + nt * 16 + r] = c[v] + b2[nt * 16 + r];
    }
}

// ODE: y0 (8 x 512) -> leaky MLP 512->256->256->512, z1 = y0 + DT*dz
__global__ __launch_bounds__(32)
void ode_kernel(const float* __restrict__ y0,
                const float* __restrict__ w1, const float* __restrict__ b1,
                const float* __restrict__ w2, const float* __restrict__ b2,
                const float* __restrict__ w3, const float* __restrict__ b3,
                float* __restrict__ z1)
{
    __shared__ float sIn[16 * 512];
    __shared__ float sA[16 * 256];
    __shared__ float sB[16 * 256];
    const int lane = threadIdx.x;
    for (int i = 0; i < 16; ++i)
        for (int j = lane; j < 512; j += 32)
            sIn[i * 512 + j] = (i < 8) ? y0[i * 512 + j] : 0.f;
    __syncthreads();
    waveGemmTile<512, 256, 2>(sIn, w1, b1, sA, lane);
    __syncthreads();
    waveGemmTile<256, 256, 2>(sA, w2, b2, sB, lane);
    __syncthreads();
    const int r = lane & 15, hi = lane >> 4;
    for (int nt = 0; nt < 32; ++nt) {
        v8f c = {};
        const float* wp = w3 + ((nt * 16 + r) * 4 + hi * 2);   // ON = 512
        for (int k4 = 0; k4 < 64; ++k4) {
            v2f b = *(const v2f*)(wp + (size_t)k4 * 32 * 64);
            v2f a = *(const v2f*)(sB + r * 256 + k4 * 4 + hi * 2);
            c = __builtin_amdgcn_wmma_f32_16x16x4_f32(false, a, false, b,
                                                      (short)0, c, false, false);
        }
#pragma unroll
        for (int v = 0; v < 8; ++v) {
            const int m = v + 8 * hi;
            if (m < 8) {
                const int col = nt * 16 + r;
                z1[m * 512 + col] = y0[m * 512 + col] + DTf * (c[v] + b3[col]);
            }
        }
    }
}

// nfl[n][l] = sum_k s[n][k] * z1[batch[n]][k][l]
__global__ __launch_bounds__(256)
void nfl_kernel(const float* __restrict__ s, const float* __restrict__ z1,
                const int* __restrict__ batch, float* __restrict__ nfl)
{
    const int idx = blockIdx.x * 256 + threadIdx.x;
    const int n = idx >> 5;
    const int l = idx & 31;
    const int b = batch[n];
    float acc = 0.f;
    for (int k = 0; k < 16; ++k)
        acc += s[(size_t)n * 16 + k] * z1[b * 512 + k * 32 + l];
    nfl[(size_t)n * 32 + l] = acc;
}

// decoder (32 nodes per wave): hd = LN(relu(nfl @ w0 + b0)); recon = hd @ w2 + b2
__global__ __launch_bounds__(32)
void dec_kernel(const float* __restrict__ nfl,
                const float* __restrict__ w0, const float* __restrict__ b0,
                const float* __restrict__ lng, const float* __restrict__ lnb,
                const float* __restrict__ w2, const float* __restrict__ b2,
                float* __restrict__ recon)
{
    extern __shared__ float smem[];   // 32*256
    const int lane = threadIdx.x;
    const int n0 = blockIdx.x * 32;
    waveGemmTile2<32, 256, 1>(nfl + (size_t)n0 * 32, nfl + (size_t)(n0 + 16) * 32,
                              w0, b0, smem, smem + 16 * 256, lane);
    __syncthreads();
    {
        float m = 0.f, sq = 0.f;
        for (int c = 0; c < 256; ++c) {
            const float v = smem[lane * 256 + c];
            m += v; sq += v * v;
        }
        m *= (1.f / 256.f);
        const float var = sq * (1.f / 256.f) - m * m;
        const float inv = rsqrtf(var + 1e-5f);
        for (int c = 0; c < 256; ++c)
            smem[lane * 256 + c] = (smem[lane * 256 + c] - m) * inv * lng[c] + lnb[c];
    }
    __syncthreads();
    float acc[8];
#pragma unroll
    for (int j = 0; j < 8; ++j) acc[j] = b2[j];
    for (int c = 0; c < 256; ++c) {
        const float hv = smem[lane * 256 + c];
#pragma unroll
        for (int j = 0; j < 8; ++j) acc[j] += hv * w2[c * 8 + j];
    }
#pragma unroll
    for (int j = 0; j < 8; ++j) recon[(size_t)(n0 + lane) * 8 + j] = acc[j];
}

__global__ void finalize_kernel(const float* Sb0, const float* Sb1, const float* A3a,
                                const float* dots, const float* entAcc,
                                const float* connAcc, float* lossOut)
{
    if (threadIdx.x != 0 || blockIdx.x != 0) return;
    float colsum[16], a1x[16], a1y[16];
    for (int k = 0; k < 16; ++k) {
        float cs = 0.f, ax = 0.f, ay = 0.f;
        for (int b = 0; b < 8; ++b) {
            cs += Sb0[b * 16 + k];
            ax += Sb1[(b * 16 + k) * 2 + 0];
            ay += Sb1[(b * 16 + k) * 2 + 1];
        }
        colsum[k] = cs; a1x[k] = ax; a1y[k] = ay;
    }
    const float entropy = -entAcc[0] / (float)Nn;
    float diversity = 0.f, pruning = 0.f, spatial = 0.f;
    for (int k = 0; k < 16; ++k) {
        const float avg = colsum[k] / (float)Nn;
        diversity += avg * logf(avg + EPSf);
        pruning += fabsf(avg);
        const float cs = colsum[k] + EPSf;
        const float mx = a1x[k] / cs, my = a1y[k] / cs;
        spatial += A3a[k] / cs - (mx * mx + my * my);
    }
    pruning *= (1.f / 16.f);
    spatial *= (1.f / 16.f);
    float sep = 0.f;
    for (int b = 0; b < 8; ++b) {
        float mbx[16], mby[16];
        for (int k = 0; k < 16; ++k) {
            const float d = Sb0[b * 16 + k] + EPSf;
            mbx[k] = Sb1[(b * 16 + k) * 2 + 0] / d;
            mby[k] = Sb1[(b * 16 + k) * 2 + 1] / d;
        }
        for (int i = 0; i < 16; ++i)
            for (int j = 0; j < 16; ++j)
                if (i != j) {
                    const float dx = mbx[i] - mbx[j], dy = mby[i] - mby[j];
                    sep += 1.f / (dx * dx + dy * dy + 1.f);
                }
    }
    sep /= (16.f * 15.f + EPSf);
    float ortho = 0.f;
    for (int t = 0; t < 256; ++t) {
        const float diag = (t % 17 == 0) ? ((float)Nn / 16.f) : 0.f;
        const float dv = dots[t] - diag;
        ortho += dv * dv;
    }
    ortho *= (1.f / 256.f);
    const float conn = connAcc[0] / ((float)Ee * 16.f);
    lossOut[0] = entropy; lossOut[1] = diversity; lossOut[2] = spatial;
    lossOut[3] = pruning; lossOut[4] = sep; lossOut[5] = ortho; lossOut[6] = conn;
}

extern "C" void kernel_launch(void* const* d_in, const int* in_sizes, int n_in,
                              void* d_out, int out_size, void* d_ws, size_t ws_size,
                              hipStream_t stream)
{
    (void)in_sizes; (void)n_in; (void)out_size; (void)ws_size;
    const float* x      = (const float*)d_in[0];   // N x F
    const int*   ei     = (const int*)d_in[1];     // 2 x E
    const int*   batch  = (const int*)d_in[2];     // N
    const float* g      = (const float*)d_in[3];   // N x K
    const float* g1w1 = (const float*)d_in[4];
    const float* g1b1 = (const float*)d_in[5];
    const float* g1w2 = (const float*)d_in[6];
    const float* g1b2 = (const float*)d_in[7];
    const float* g1w3 = (const float*)d_in[8];
    const float* g1b3 = (const float*)d_in[9];
    const float* ln1g = (const float*)d_in[10];
    const float* ln1b = (const float*)d_in[11];
    const float* g2w1 = (const float*)d_in[12];
    const float* g2b1 = (const float*)d_in[13];
    const float* g2w2 = (const float*)d_in[14];
    const float* g2b2 = (const float*)d_in[15];
    const float* g2w3 = (const float*)d_in[16];
    const float* g2b3 = (const float*)d_in[17];
    const float* ln2g = (const float*)d_in[18];
    const float* ln2b = (const float*)d_in[19];
    const float* pw1  = (const float*)d_in[20];
    const float* pb1  = (const float*)d_in[21];
    const float* pw2  = (const float*)d_in[22];
    const float* pb2  = (const float*)d_in[23];
    const float* pscl = (const float*)d_in[24];
    const float* ew1  = (const float*)d_in[25];
    const float* eb1  = (const float*)d_in[26];
    const float* ew2  = (const float*)d_in[27];
    const float* eb2  = (const float*)d_in[28];
    const float* ow1  = (const float*)d_in[29];
    const float* ob1  = (const float*)d_in[30];
    const float* ow2  = (const float*)d_in[31];
    const float* ob2  = (const float*)d_in[32];
    const float* ow3  = (const float*)d_in[33];
    const float* ob3  = (const float*)d_in[34];
    const float* dw0  = (const float*)d_in[35];
    const float* db0  = (const float*)d_in[36];
    const float* dlng = (const float*)d_in[37];
    const float* dlnb = (const float*)d_in[38];
    const float* dw2  = (const float*)d_in[39];
    const float* db2  = (const float*)d_in[40];

    float* W = (float*)d_ws;
    size_t o = 0;
    float* accA   = W + o; o += (size_t)Nn * Cc;
    float* hBuf   = W + o; o += (size_t)Nn * Cc;
    float* sBuf   = W + o; o += (size_t)Nn * Kk;
    float* nflBuf = W + o; o += (size_t)Nn * Ll;
    float* pooled = W + o; o += Bb * Kk * Cc;
    float* latent = W + o; o += Bb * Kk * Ll;
    float* z1buf  = W + o; o += Bb * Kk * Ll;
    float* stats  = W + o;
    float* Sb0    = stats;          // 128
    float* Sb1    = stats + 128;    // 256
    float* A3a    = stats + 384;    // 16
    float* dots   = stats + 400;    // 256
    float* entAcc = stats + 656;    // 1
    float* connAcc= stats + 657;    // 1
    o += 1024;
    int* startB = (int*)(W + o);
    int* endB   = startB + 8;
    o += 16;
    // repacked weights (fragment-contiguous)
    float* rg1w1 = W + o; o += 16 * 256;
    float* rg1w2 = W + o; o += 256 * 256;
    float* rg1w3 = W + o; o += 256 * 256;
    float* rg2w1 = W + o; o += 512 * 256;
    float* rg2w2 = W + o; o += 256 * 256;
    float* rg2w3 = W + o; o += 256 * 256;
    float* rpw1  = W + o; o += 256 * 256;
    float* rpw2  = W + o; o += 256 * 16;
    float* rew1  = W + o; o += 256 * 256;
    float* rew2  = W + o; o += 256 * 32;
    float* row1  = W + o; o += 512 * 256;
    float* row2  = W + o; o += 256 * 256;
    float* row3  = W + o; o += 256 * 512;
    float* rdw0  = W + o; o += 32 * 256;

    // ---- weight repack (fragment order, one b64 per B fragment) ----
    repack_kernel<16, 256><<<16 * 256 / 256, 256, 0, stream>>>(g1w1, rg1w1);
    repack_kernel<256, 256><<<256 * 256 / 256, 256, 0, stream>>>(g1w2, rg1w2);
    repack_kernel<256, 256><<<256 * 256 / 256, 256, 0, stream>>>(g1w3, rg1w3);
    repack_kernel<512, 256><<<512 * 256 / 256, 256, 0, stream>>>(g2w1, rg2w1);
    repack_kernel<256, 256><<<256 * 256 / 256, 256, 0, stream>>>(g2w2, rg2w2);
    repack_kernel<256, 256><<<256 * 256 / 256, 256, 0, stream>>>(g2w3, rg2w3);
    repack_kernel<256, 256><<<256 * 256 / 256, 256, 0, stream>>>(pw1, rpw1);
    repack_kernel<256, 16><<<256 * 16 / 256, 256, 0, stream>>>(pw2, rpw2);
    repack_kernel<256, 256><<<256 * 256 / 256, 256, 0, stream>>>(ew1, rew1);
    repack_kernel<256, 32><<<256 * 32 / 256, 256, 0, stream>>>(ew2, rew2);
    repack_kernel<512, 256><<<512 * 256 / 256, 256, 0, stream>>>(ow1, row1);
    repack_kernel<256, 256><<<256 * 256 / 256, 256, 0, stream>>>(ow2, row2);
    repack_kernel<256, 512><<<256 * 512 / 256, 256, 0, stream>>>(ow3, row3);
    repack_kernel<32, 256><<<32 * 256 / 256, 256, 0, stream>>>(dw0, rdw0);

    const size_t shmGnn1 = (size_t)(32 * 16 + 32 * 256 + 32 * 256) * sizeof(float); // 66 KB
    const size_t shmGnn2 = (size_t)(32 * 512 + 32 * 256) * sizeof(float);           // 96 KB
    const size_t shmPool = (size_t)(32 * 256) * sizeof(float);                      // 32 KB
    const size_t shmDec  = (size_t)(32 * 256) * sizeof(float);                      // 32 KB

    // ---- GNN layer 1 ----
    hipMemsetAsync(accA, 0, (size_t)Nn * Cc * sizeof(float), stream);
    gnn_edge_kernel<16><<<Ee / 32, 32, shmGnn1, stream>>>(x, ei, rg1w1, g1b1, rg1w2, g1b2,
                                                          rg1w3, g1b3, accA);
    relu_ln_kernel<<<Nn, 32, 0, stream>>>(accA, ln1g, ln1b, hBuf);

    // ---- GNN layer 2 ----
    hipMemsetAsync(accA, 0, (size_t)Nn * Cc * sizeof(float), stream);
    gnn_edge_kernel<512><<<Ee / 32, 32, shmGnn2, stream>>>(hBuf, ei, rg2w1, g2b1, rg2w2, g2b2,
                                                           rg2w3, g2b3, accA);
    relu_ln_kernel<<<Nn, 32, 0, stream>>>(accA, ln2g, ln2b, hBuf);

    // ---- pool + softmax ----
    pool_softmax_kernel<<<Nn / 32, 32, shmPool, stream>>>(hBuf, rpw1, pb1, rpw2, pb2,
                                                          pscl, g, sBuf);

    // ---- statistics / losses accumulation ----
    hipMemsetAsync(stats, 0, 1024 * sizeof(float), stream);
    range_init_kernel<<<1, 32, 0, stream>>>(startB, endB);
    range_kernel<<<Nn / 256, 256, 0, stream>>>(batch, startB, endB);
    stats_kernel<<<Nn / 256, 256, 0, stream>>>(sBuf, x, batch, Sb0, Sb1, A3a, dots, entAcc);
    conn_kernel<<<Ee / 256, 256, 0, stream>>>(sBuf, ei, connAcc);

    // ---- pooled = s_b^T @ h_b per batch ----
    pooled_kernel<<<Bb, 32, 0, stream>>>(sBuf, hBuf, startB, endB, pooled);

    // ---- encoder out / ODE / decoder ----
    encout_kernel<<<(Bb * Kk) / 16, 32, 0, stream>>>(pooled, rew1, eb1, rew2, eb2, latent);
    ode_kernel<<<1, 32, 0, stream>>>(latent, row1, ob1, row2, ob2, row3, ob3, z1buf);
    nfl_kernel<<<(Nn * Ll) / 256, 256, 0, stream>>>(sBuf, z1buf, batch, nflBuf);
    dec_kernel<<<Nn / 32, 32, shmDec, stream>>>(nflBuf, dw0, db0, dlng, dlnb, dw2, db2,
                                                (float*)d_out);

    // ---- final scalar losses ----
    finalize_kernel<<<1, 32, 0, stream>>>(Sb0, Sb1, A3a, dots, entAcc, connAcc,
                                          (float*)d_out + (size_t)Nn * Ff);
}